// NeighborEmbedding_39651138076968
// MI455X (gfx1250) — compile-verified
//
#include <hip/hip_runtime.h>

typedef __attribute__((ext_vector_type(2))) float v2f;
typedef __attribute__((ext_vector_type(8))) float v8f;

#define N_NODES   50000
#define N_EDGES   600000
#define N_ELEM    100
#define N_RBF     50
#define DIMS      128
#define CUTOFF_F  5.0f

// ---------------------------------------------------------------------------
// Zero the scatter accumulator (must be re-done every call: determinism).
// ---------------------------------------------------------------------------
__global__ void zero_kernel(float* __restrict__ p, int n) {
  int i = blockIdx.x * blockDim.x + threadIdx.x;
  if (i < n) p[i] = 0.f;
}

// ---------------------------------------------------------------------------
// Fused edge kernel:
//   x_edges = (rbf @ W_e.T + b_e) * cutoff(dist)      [WMMA f32 16x16x4]
//   msg     = emb[node_type[src]] * x_edges
//   atomicAdd into xint[dst]
// One wave handles 16 edges x 128 dims (8 C-tiles). K=50 padded to 52.
// A-fragment lane layout (32-bit 16x4):  lane<16 -> M=lane, K={k0,k0+1};
//                                        lane>=16 -> M=lane-16, K={k0+2,k0+3}.
// B-fragment (4x16):                     lane<16 -> N=lane, K={k0,k0+1};
//                                        lane>=16 -> N=lane-16, K={k0+2,k0+3}.
// ---------------------------------------------------------------------------
__global__ __launch_bounds__(256) void edge_kernel(
    const int*   __restrict__ node_type,
    const int*   __restrict__ src,
    const int*   __restrict__ dst,
    const float* __restrict__ rbf,
    const float* __restrict__ dist,
    const float* __restrict__ emb,
    const float* __restrict__ W_e,
    const float* __restrict__ b_e,
    float*       __restrict__ xint)
{
  // W_e staged as [n][k], stride 52 (k=50,51 zero pad). Stride 52 -> banks
  // (4*i + kb) mod 64: conflict-free for both 16-lane halves.
  __shared__ float sWe[DIMS * 52];
  __shared__ float sBe[DIMS];

  const int tid = threadIdx.x;
  for (int i = tid; i < DIMS * 52; i += 256) {
    int r = i / 52, c = i - r * 52;
    sWe[i] = (c < N_RBF) ? W_e[r * N_RBF + c] : 0.f;
  }
  if (tid < DIMS) sBe[tid] = b_e[tid];
  __syncthreads();

  const int lane = tid & 31;
  const int wave = tid >> 5;
  const int tile = blockIdx.x * 8 + wave;        // 16-edge tile
  if (tile >= N_EDGES / 16) return;              // wave-uniform guard
  const int e0 = tile * 16;

  // Per-edge metadata; both lane halves redundantly compute edge (lane&15).
  const int m    = lane & 15;
  const int em   = e0 + m;
  const float dd = dist[em];
  float wgt = 0.f;
  if (dd < CUTOFF_F)
    wgt = 0.5f * (__cosf(dd * (3.14159265358979f / CUTOFF_F)) + 1.f);
  const int ty = node_type[src[em]];
  const int dn = dst[em];

  v8f acc[8] = {};

  const int koff = (lane >> 4) << 1;             // 0 or 2
  const float* arow = rbf + (size_t)em * N_RBF;  // row of A (this lane's edge)

  #pragma unroll
  for (int kk = 0; kk < 13; ++kk) {
    const int kb = kk * 4 + koff;
    v2f a;
    if (kk < 12) {
      a = *(const v2f*)(arow + kb);              // 8B aligned (200B row stride)
    } else {                                     // K tail: 48,49 valid; 50,51 -> 0
      a.x = (kb     < N_RBF) ? arow[kb]     : 0.f;
      a.y = (kb + 1 < N_RBF) ? arow[kb + 1] : 0.f;
    }
    #pragma unroll
    for (int nt = 0; nt < 8; ++nt) {
      const int n = nt * 16 + m;
      const v2f b = *(const v2f*)(sWe + n * 52 + kb);   // pad cols are zero
      acc[nt] = __builtin_amdgcn_wmma_f32_16x16x4_f32(
          false, a, false, b, (short)0, acc[nt], false, false);
    }
  }

  // Epilogue: C/D layout puts row M = r + 8*(lane>=16) in acc[.][r].
  #pragma unroll
  for (int r = 0; r < 8; ++r) {
    const int   mm = r + ((lane >> 4) << 3);
    const float wm = __shfl(wgt, mm, 32);
    const int   tm = __shfl(ty,  mm, 32);
    const int   dm = __shfl(dn,  mm, 32);
    if (wm > 0.f) {                              // beyond cutoff contributes 0
      const float* erow = emb  + (size_t)tm * DIMS;
      float*       orow = xint + (size_t)dm * DIMS;
      #pragma unroll
      for (int nt = 0; nt < 8; ++nt) {
        const int n = nt * 16 + m;
        const float v = (acc[nt][r] + sBe[n]) * wm;
        atomicAdd(orow + n, v * erow[n]);
      }
    }
  }
}

// ---------------------------------------------------------------------------
// Combine kernel: out = concat(x_nodes, xint) @ W_c.T + b_c.
// Wave = 16 rows, block = 128 rows. W_c streamed through LDS in four 64-wide
// K-slabs ([n][k] stride 68 -> conflict-free) so each block reads W_c once.
// Safe in-place when xint == out: a wave reads its 16 rows fully (k-loop)
// before writing them, and each row is owned by exactly one wave.
// ---------------------------------------------------------------------------
__global__ __launch_bounds__(256) void combine_kernel(
    const float* __restrict__ x_nodes,
    const float* __restrict__ xint,
    const float* __restrict__ W_c,
    const float* __restrict__ b_c,
    float*       __restrict__ out)
{
  __shared__ float sB[DIMS * 68];                // 34.8 KB

  const int tid  = threadIdx.x;
  const int lane = tid & 31;
  const int wave = tid >> 5;
  const int tile = blockIdx.x * 8 + wave;
  const bool active = tile < (N_NODES / 16);     // wave-uniform
  const int row0 = tile * 16;
  const int m    = lane & 15;
  const int koff = (lane >> 4) << 1;

  v8f acc[8] = {};

  for (int ks = 0; ks < 4; ++ks) {
    const int k0 = ks * 64;
    __syncthreads();                             // protect previous slab
    for (int i = tid; i < DIMS * 64; i += 256) {
      const int n = i >> 6, c = i & 63;
      sB[n * 68 + c] = W_c[n * 256 + k0 + c];    // B[k][n] = W_c[n][k]
    }
    __syncthreads();

    if (active) {
      const float* Abase = (k0 < DIMS)
          ? (x_nodes + (size_t)(row0 + m) * DIMS + k0)
          : (xint    + (size_t)(row0 + m) * DIMS + (k0 - DIMS));
      #pragma unroll
      for (int kk = 0; kk < 16; ++kk) {
        const int kb = kk * 4 + koff;            // 0..62, even
        const v2f a = *(const v2f*)(Abase + kb);
        #pragma unroll
        for (int nt = 0; nt < 8; ++nt) {
          const int n = nt * 16 + m;
          const v2f b = *(const v2f*)(sB + n * 68 + kb);
          acc[nt] = __builtin_amdgcn_wmma_f32_16x16x4_f32(
              false, a, false, b, (short)0, acc[nt], false, false);
        }
      }
    }
  }

  if (active) {
    #pragma unroll
    for (int r = 0; r < 8; ++r) {
      const int mm = r + ((lane >> 4) << 3);
      float* orow = out + (size_t)(row0 + mm) * DIMS;
      #pragma unroll
      for (int nt = 0; nt < 8; ++nt) {
        const int n = nt * 16 + m;
        orow[n] = acc[nt][r] + b_c[n];
      }
    }
  }
}

// ---------------------------------------------------------------------------
extern "C" void kernel_launch(void* const* d_in, const int* in_sizes, int n_in,
                              void* d_out, int out_size, void* d_ws, size_t ws_size,
                              hipStream_t stream) {
  const int*   node_type = (const int*)  d_in[0];
  const float* x_nodes   = (const float*)d_in[1];
  const int*   src       = (const int*)  d_in[2];
  const int*   dst       = (const int*)  d_in[3];
  const float* rbf       = (const float*)d_in[4];
  const float* dist      = (const float*)d_in[5];
  const float* emb       = (const float*)d_in[6];
  const float* W_e       = (const float*)d_in[7];
  const float* b_e       = (const float*)d_in[8];
  const float* W_c       = (const float*)d_in[9];
  const float* b_c       = (const float*)d_in[10];
  float* out = (float*)d_out;

  // Scatter accumulator: workspace if big enough, else compute in-place in out.
  const size_t need = (size_t)N_NODES * DIMS * sizeof(float);
  float* xint = (ws_size >= need) ? (float*)d_ws : out;

  const int zn = N_NODES * DIMS;
  zero_kernel<<<(zn + 255) / 256, 256, 0, stream>>>(xint, zn);

  const int etiles = N_EDGES / 16;               // 37500
  edge_kernel<<<(etiles + 7) / 8, 256, 0, stream>>>(
      node_type, src, dst, rbf, dist, emb, W_e, b_e, xint);

  const int ntiles = N_NODES / 16;               // 3125
  combine_kernel<<<(ntiles + 7) / 8, 256, 0, stream>>>(
      x_nodes, xint, W_c, b_c, out);
}